// NearestMatcher_90288802497279
// MI455X (gfx1250) — compile-verified
//
#include <hip/hip_runtime.h>
#include <hip/hip_bf16.h>

// NearestMatcher for MI455X (gfx1250, wave32).
// Memory-bound target (~300MB traffic => ~13us at 23.3 TB/s). CDNA5 async
// global->LDS copies (ASYNCcnt path) stage each block's inputs; one row per
// thread computes out of LDS. Dedup does pairwise stable-rank comparison with
// bool (lane-mask) accumulators so the combine logic lands on the SALU pipe
// and co-issues with VALU, keeping VALU below the HBM roof.

#define BLOCK 256
#define NQ 16
#define NT 8
#define BIGSENT 1000000.0f

// LDS layout (byte offsets; padded rows to avoid bank conflicts, 16B aligned)
#define ROWQ_BYTES 80   // 16 floats padded to 20
#define ROWT_BYTES 48   // 8 floats padded to 12
#define LOG_OFF 0
#define DISP_OFF (BLOCK * ROWQ_BYTES)          // 20480
#define TGT_OFF  (2 * BLOCK * ROWQ_BYTES)      // 40960
#define SMEM_BYTES (TGT_OFF + BLOCK * ROWT_BYTES)  // 53248

__device__ __forceinline__ void async_load16(unsigned lds_off, const void* gaddr) {
    // GLOBAL_LOAD_ASYNC_TO_LDS_B128, GV mode: vdst = LDS byte address (VGPR),
    // vaddr = 64-bit global address (VGPR pair). Tracked by ASYNCcnt.
    asm volatile("global_load_async_to_lds_b128 %0, %1, off"
                 :: "v"(lds_off), "v"((unsigned long long)gaddr)
                 : "memory");
}

__global__ __launch_bounds__(BLOCK)
void NearestMatcher_90288802497279_kernel(const float* __restrict__ pred_logits,
                                          const float* __restrict__ pred_disp,
                                          const float* __restrict__ targets,
                                          float* __restrict__ out_indices,
                                          float* __restrict__ out_labels,
                                          int n_rows) {
    __shared__ __align__(16) unsigned char smem[SMEM_BYTES];

    // Escape the LDS buffer into asm so the asm "memory" clobbers are known to
    // (asynchronously) write it; otherwise LLVM folds the never-stored smem
    // loads to undef and deletes the whole compute phase.
    asm volatile("" :: "v"((void*)smem) : "memory");

    const int tid = threadIdx.x;
    const long long block0 = (long long)blockIdx.x * BLOCK;
    const int rem_rows = min(BLOCK, n_rows - (int)min((long long)n_rows, block0));

    const unsigned char* glog = (const unsigned char*)(pred_logits + block0 * NQ);
    const unsigned char* gdis = (const unsigned char*)(pred_disp   + block0 * NQ);
    const unsigned char* gtgt = (const unsigned char*)(targets     + block0 * NT);

    // ---- Stage inputs into LDS with async copies (16B per transfer) ----
    const int rowq = tid >> 2, chq = tid & 3;   // for 16-float rows (4 xfers/row)
    const int rowt = tid >> 1, cht = tid & 1;   // for 8-float rows (2 xfers/row)
    if (rem_rows == BLOCK) {
        // Hot path: full block, no guards -> no exec-mask churn.
#pragma unroll
        for (int i = 0; i < 4; ++i) {  // 64 rows per step
            async_load16((unsigned)(LOG_OFF + (rowq + i * 64) * ROWQ_BYTES + chq * 16),
                         glog + (tid + i * BLOCK) * 16);
            async_load16((unsigned)(DISP_OFF + (rowq + i * 64) * ROWQ_BYTES + chq * 16),
                         gdis + (tid + i * BLOCK) * 16);
        }
#pragma unroll
        for (int i = 0; i < 2; ++i) {  // 128 rows per step
            async_load16((unsigned)(TGT_OFF + (rowt + i * 128) * ROWT_BYTES + cht * 16),
                         gtgt + (tid + i * BLOCK) * 16);
        }
    } else {
        // Tail block: guard each transfer.
#pragma unroll
        for (int i = 0; i < 4; ++i) {
            int f = tid + i * BLOCK;
            if (f < rem_rows * 4) {
                async_load16((unsigned)(LOG_OFF + (f >> 2) * ROWQ_BYTES + (f & 3) * 16),
                             glog + f * 16);
                async_load16((unsigned)(DISP_OFF + (f >> 2) * ROWQ_BYTES + (f & 3) * 16),
                             gdis + f * 16);
            }
        }
#pragma unroll
        for (int i = 0; i < 2; ++i) {
            int f = tid + i * BLOCK;
            if (f < rem_rows * 2) {
                async_load16((unsigned)(TGT_OFF + (f >> 1) * ROWT_BYTES + (f & 1) * 16),
                             gtgt + f * 16);
            }
        }
    }
    asm volatile("s_wait_asynccnt 0" ::: "memory");
    __syncthreads();

    const long long row = block0 + tid;
    if (row >= n_rows) return;

    // ---- Read this thread's row from LDS (padded strides) ----
    float lg[NQ], dv[NQ], td[NT];
    {
        const float4* lp = (const float4*)(&smem[LOG_OFF + tid * ROWQ_BYTES]);
        const float4* dp = (const float4*)(&smem[DISP_OFF + tid * ROWQ_BYTES]);
#pragma unroll
        for (int i = 0; i < 4; ++i) {
            float4 a = lp[i];
            lg[4*i+0] = a.x; lg[4*i+1] = a.y; lg[4*i+2] = a.z; lg[4*i+3] = a.w;
            float4 b = dp[i];
            dv[4*i+0] = b.x; dv[4*i+1] = b.y; dv[4*i+2] = b.z; dv[4*i+3] = b.w;
        }
        const float4* tp = (const float4*)(&smem[TGT_OFF + tid * ROWT_BYTES]);
        float4 t0 = tp[0], t1 = tp[1];
        td[0] = t0.x; td[1] = t0.y; td[2] = t0.z; td[3] = t0.w;
        td[4] = t1.x; td[5] = t1.y; td[6] = t1.z; td[7] = t1.w;
    }

    // sentinel for null disparity targets
#pragma unroll
    for (int t = 0; t < NT; ++t)
        td[t] = (td[t] == 0.0f) ? BIGSENT : td[t];

    // ---- Per-query nearest target (first-occurrence argmin) + total cost ----
    float C[NQ];
    int   bidx[NQ];
#pragma unroll
    for (int n = 0; n < NQ; ++n) {
        float d = dv[n];
        float best = fabsf(d - td[0]);
        int bi = 0;
#pragma unroll
        for (int t = 1; t < NT; ++t) {
            float v = fabsf(d - td[t]);       // |..| folds to a source modifier
            bool lt = v < best;               // strict < keeps first occurrence
            bi = lt ? t : bi;
            best = lt ? v : best;
        }
        bidx[n] = bi;
        // C = COST_CLASS * (-sigmoid(logit)) + COST_DISP * min-dist
        C[n] = best - 1.0f / (1.0f + __expf(-lg[n]));
    }

    // ---- Dedup via pairwise stable rank (== stable argsort + first-occ).
    // For i<j: i precedes j iff C[i] <= C[j] (tie -> lower index). Keep the
    // accumulators as bools: lane-mask logic lowers to s_and/s_or/s_andn2 on
    // the SALU pipe (co-issues with VALU); only the two v_cmp per pair hit
    // the VALU. (C is never NaN here, so "j precedes i" == !le.) ----
    bool killed[NQ];
#pragma unroll
    for (int n = 0; n < NQ; ++n) killed[n] = false;
#pragma unroll
    for (int i = 0; i < NQ; ++i) {
#pragma unroll
        for (int j = i + 1; j < NQ; ++j) {
            bool e  = (bidx[i] == bidx[j]);
            bool le = (C[i] <= C[j]);
            killed[j] = killed[j] || (e && le);
            killed[i] = killed[i] || (e && !le);
        }
    }

    // ---- Store outputs (indices as float, then labels), float4-vectorized ----
    const long long obase = row * NQ;
    float4* oi = (float4*)(out_indices + obase);
    float4* ol = (float4*)(out_labels + obase);
#pragma unroll
    for (int i = 0; i < 4; ++i) {
        oi[i] = make_float4((float)bidx[4*i+0], (float)bidx[4*i+1],
                            (float)bidx[4*i+2], (float)bidx[4*i+3]);
        ol[i] = make_float4(killed[4*i+0] ? 0.0f : 1.0f,
                            killed[4*i+1] ? 0.0f : 1.0f,
                            killed[4*i+2] ? 0.0f : 1.0f,
                            killed[4*i+3] ? 0.0f : 1.0f);
    }
}

extern "C" void kernel_launch(void* const* d_in, const int* in_sizes, int n_in,
                              void* d_out, int out_size, void* d_ws, size_t ws_size,
                              hipStream_t stream) {
    const float* pred_logits = (const float*)d_in[0];  // [B,16]
    const float* pred_disp   = (const float*)d_in[1];  // [B,16]
    const float* targets     = (const float*)d_in[2];  // [B,8]

    const int BN = in_sizes[0];        // B * 16
    const int n_rows = BN / NQ;        // B

    float* out_indices = (float*)d_out;        // first B*16 elements
    float* out_labels  = (float*)d_out + BN;   // next  B*16 elements

    const int grid = (n_rows + BLOCK - 1) / BLOCK;
    NearestMatcher_90288802497279_kernel<<<grid, BLOCK, 0, stream>>>(
        pred_logits, pred_disp, targets, out_indices, out_labels, n_rows);
}